// Linear_GCN_75488345194747
// MI455X (gfx1250) — compile-verified
//
#include <hip/hip_runtime.h>
#include <hip/hip_bf16.h>

// ---------------------------------------------------------------------------
// GCN forward for MI455X (gfx1250, wave32).
//   h       = relu(x @ W1 + b1)        x:[N,512] W1:[512,128]
//   support = h @ W2                   W2:[128,64]   -> workspace
//   out     = segment_sum(support[src]*w, dst) + b2  -> d_out [N,64]
// GEMMs use V_WMMA_F32_16X16X4_F32 (fp32-exact matrix path on CDNA5).
// ---------------------------------------------------------------------------

typedef float v2f __attribute__((ext_vector_type(2)));
typedef float v8f __attribute__((ext_vector_type(8)));

#define NFEAT  512
#define NHID   128
#define NCLASS 64
#define XS     516   // LDS row stride for x tile (pad: 512 % 64banks == 0)
#define HS     132   // LDS row stride for h tile (pad: 128 % 64banks == 0)

// One wave (32 lanes) per 16-row tile of x.
__global__ __launch_bounds__(32)
void gcn_fused_gemm(const float* __restrict__ x,
                    const float* __restrict__ W1,
                    const float* __restrict__ b1,
                    const float* __restrict__ W2,
                    float* __restrict__ support,
                    int n_nodes)
{
    __shared__ float xs[16 * XS];   // 16 x 512 fp32 tile of x (padded)
    __shared__ float hs[16 * HS];   // 16 x 128 fp32 tile of h (padded)

    const int lane = threadIdx.x & 31;
    const int half = lane >> 4;     // 0: lanes 0-15, 1: lanes 16-31
    const int l15  = lane & 15;
    const int row0 = blockIdx.x * 16;

    // ---- stage x tile into LDS, float4-coalesced -------------------------
    for (int q = lane; q < 16 * (NFEAT / 4); q += 32) {
        const int r   = q >> 7;            // q / 128
        const int c   = (q & 127) << 2;    // (q % 128) * 4
        const int row = row0 + r;
        float4 v = make_float4(0.f, 0.f, 0.f, 0.f);
        if (row < n_nodes)
            v = *(const float4*)(x + (size_t)row * NFEAT + c);
        float* p = xs + r * XS + c;
        p[0] = v.x; p[1] = v.y; p[2] = v.z; p[3] = v.w;
    }
    __syncthreads();

    // ---- GEMM1: h[16,128] = relu(xtile @ W1 + b1), fp32 WMMA -------------
    // A frag (16x4): lanes 0-15 hold K=k,k+1 ; lanes 16-31 hold K=k+2,k+3.
    // B frag (4x16): N striped across lanes, same K split.
    for (int nt = 0; nt < NHID / 16; ++nt) {
        v8f acc = {};
        for (int k = 0; k < NFEAT; k += 4) {
            const int kk = k + 2 * half;
            v2f a, b;
            a.x = xs[l15 * XS + kk];
            a.y = xs[l15 * XS + kk + 1];
            b.x = W1[(size_t)kk       * NHID + nt * 16 + l15];
            b.y = W1[(size_t)(kk + 1) * NHID + nt * 16 + l15];
            acc = __builtin_amdgcn_wmma_f32_16x16x4_f32(
                false, a, false, b, (short)0, acc, false, false);
        }
        // C/D layout: VGPR r holds row (r + 8*half), column l15 of the tile.
        const float bias = b1[nt * 16 + l15];
        #pragma unroll
        for (int r = 0; r < 8; ++r) {
            const int m = r + 8 * half;
            float v = acc[r] + bias;
            hs[m * HS + nt * 16 + l15] = v > 0.f ? v : 0.f;
        }
    }
    __syncthreads();

    // ---- GEMM2: support[16,64] = htile @ W2, fp32 WMMA -------------------
    for (int ct = 0; ct < NCLASS / 16; ++ct) {
        v8f acc = {};
        for (int k = 0; k < NHID; k += 4) {
            const int kk = k + 2 * half;
            v2f a, b;
            a.x = hs[l15 * HS + kk];
            a.y = hs[l15 * HS + kk + 1];
            b.x = W2[(size_t)kk       * NCLASS + ct * 16 + l15];
            b.y = W2[(size_t)(kk + 1) * NCLASS + ct * 16 + l15];
            acc = __builtin_amdgcn_wmma_f32_16x16x4_f32(
                false, a, false, b, (short)0, acc, false, false);
        }
        #pragma unroll
        for (int r = 0; r < 8; ++r) {
            const int m   = r + 8 * half;
            const int row = row0 + m;
            if (row < n_nodes)
                support[(size_t)row * NCLASS + ct * 16 + l15] = acc[r];
        }
    }
}

// out[n,c] = b2[c]  (d_out is poisoned by the harness; must fully init)
__global__ __launch_bounds__(256)
void gcn_init_out(float* __restrict__ out, const float* __restrict__ b2,
                  int total)
{
    const int t = blockIdx.x * blockDim.x + threadIdx.x;
    if (t < total) out[t] = b2[t & (NCLASS - 1)];
}

// 16 threads per edge; each thread handles a float4 chunk of the 64-wide row.
// Gather from L2-resident support, scale, native fp32 atomic scatter-add.
__global__ __launch_bounds__(256)
void gcn_scatter(const float* __restrict__ support,
                 const long long* __restrict__ src,
                 const long long* __restrict__ dst,
                 const float* __restrict__ ew,
                 float* __restrict__ out,
                 long long n_edges)
{
    const long long t = (long long)blockIdx.x * blockDim.x + threadIdx.x;
    const long long e = t >> 4;
    if (e >= n_edges) return;
    const int c0 = (int)(t & 15) << 2;         // 0,4,...,60

    const long long s = src[e];
    const long long d = dst[e];
    const float     w = ew[e];

    const float4 v = *(const float4*)(support + s * NCLASS + c0);
    float* o = out + d * NCLASS + c0;
    unsafeAtomicAdd(o + 0, v.x * w);           // -> global_atomic_add_f32
    unsafeAtomicAdd(o + 1, v.y * w);
    unsafeAtomicAdd(o + 2, v.z * w);
    unsafeAtomicAdd(o + 3, v.w * w);
}

extern "C" void kernel_launch(void* const* d_in, const int* in_sizes, int n_in,
                              void* d_out, int out_size, void* d_ws, size_t ws_size,
                              hipStream_t stream)
{
    const float*     x  = (const float*)d_in[0];      // [N,512]
    const float*     W1 = (const float*)d_in[1];      // [512,128]
    const float*     b1 = (const float*)d_in[2];      // [128]
    const float*     W2 = (const float*)d_in[3];      // [128,64]
    const float*     b2 = (const float*)d_in[4];      // [64]
    const long long* ei = (const long long*)d_in[5];  // [2,E] int64
    const float*     ew = (const float*)d_in[6];      // [E]

    const int       n_nodes = in_sizes[0] / NFEAT;
    const long long n_edges = (long long)in_sizes[5] / 2;

    float* support = (float*)d_ws;                    // [N,64] scratch
    float* out     = (float*)d_out;                   // [N,64]

    // 1) fused GEMM1(+bias+relu) and GEMM2 -> support
    gcn_fused_gemm<<<(n_nodes + 15) / 16, 32, 0, stream>>>(
        x, W1, b1, W2, support, n_nodes);

    // 2) out = b2 broadcast
    const int total = n_nodes * NCLASS;
    gcn_init_out<<<(total + 255) / 256, 256, 0, stream>>>(out, b2, total);

    // 3) edge gather/scale/scatter-add
    const long long threads = n_edges * 16;
    gcn_scatter<<<(unsigned)((threads + 255) / 256), 256, 0, stream>>>(
        support, ei, ei + n_edges, ew, out, n_edges);
}